// RandomSegmentMask_77781857730967
// MI455X (gfx1250) — compile-verified
//
#include <hip/hip_runtime.h>
#include <hip/hip_bf16.h>

#define T_LEN 4096
#define BLK   256
#define EPT   16      // elements per thread: 256*16 = 4096 = T_LEN

__global__ __launch_bounds__(BLK) void RandomSegmentMask_kernel(
    const float* __restrict__ x,          // [rows * T]
    const int*   __restrict__ idx_mask,   // [rows * T] (bool as int32)
    const int*   __restrict__ mask_size,  // [S]
    const int*   __restrict__ mask_value, // [1]
    float*       __restrict__ out,        // [rows * T]
    float*       __restrict__ mask_out,   // [rows * T] (bool as 0.0/1.0)
    int S)
{
    __shared__ float xs[T_LEN];          // staged x row (16 KB)
    __shared__ int   wtot[BLK / 32];     // per-wave scan totals

    const int tid = threadIdx.x;
    const int row = blockIdx.x;                 // [0, B*S)
    const int s   = row % S;
    const long long rowbase = (long long)row * T_LEN;

    // ---- async-stage the x row into LDS (overlaps with the scan below) ----
    // chunk layout: chunk c (16B) <-> floats [4c, 4c+4); per issue j the 32
    // lanes of a wave touch a contiguous 512B span (perfectly coalesced).
    const unsigned lds_base = (unsigned)(size_t)(&xs[0]);   // low 32b of flat = LDS offset
    const unsigned long long grow = (unsigned long long)(size_t)(x + rowbase);
#pragma unroll
    for (int j = 0; j < 4; ++j) {
        const int chunk = j * BLK + tid;                    // 1024 chunks total
        const unsigned laddr = lds_base + (unsigned)chunk * 16u;
        const unsigned long long gaddr = grow + (unsigned long long)chunk * 16ull;
        asm volatile("global_load_async_to_lds_b128 %0, %1, off"
                     :: "v"(laddr), "v"(gaddr)
                     : "memory");
    }

    // ---- load this thread's 16 idx values, local "last start" scan ----
    const int t0 = tid * EPT;
    const int4* irow = (const int4*)(idx_mask + rowbase + t0);
    int last = -1;
    int loc[EPT];
#pragma unroll
    for (int j = 0; j < 4; ++j) {
        const int4 v = irow[j];
        const int base = t0 + j * 4;
        last = v.x ? (base + 0) : last; loc[j * 4 + 0] = last;
        last = v.y ? (base + 1) : last; loc[j * 4 + 1] = last;
        last = v.z ? (base + 2) : last; loc[j * 4 + 2] = last;
        last = v.w ? (base + 3) : last; loc[j * 4 + 3] = last;
    }

    // ---- wave32 inclusive max-scan of per-thread last-start ----
    const int lane = tid & 31;
    int scan = last;
#pragma unroll
    for (int d = 1; d < 32; d <<= 1) {
        const int up = __shfl_up(scan, d, 32);
        if (lane >= d) scan = (up > scan) ? up : scan;
    }
    int lane_ex = __shfl_up(scan, 1, 32);       // exclusive within wave
    if (lane == 0) lane_ex = -1;

    const int wave = tid >> 5;
    if (lane == 31) wtot[wave] = scan;
    __syncthreads();
    int wprev = -1;
    for (int w = 0; w < wave; ++w) {
        const int ww = wtot[w];
        wprev = (ww > wprev) ? ww : wprev;
    }
    const int tex = (lane_ex > wprev) ? lane_ex : wprev;   // exclusive over all earlier t

    const int   m    = mask_size[s];
    const float mval = (float)mask_value[0];

    // ---- wait for async x staging, then fuse mask + select + store ----
    asm volatile("s_wait_asynccnt 0x0" ::: "memory");
    __syncthreads();

    float4* orow = (float4*)(out      + rowbase + t0);
    float4* mrow = (float4*)(mask_out + rowbase + t0);
#pragma unroll
    for (int j = 0; j < 4; ++j) {
        const float4 xv = *(const float4*)(&xs[t0 + j * 4]);
        const float xa[4] = {xv.x, xv.y, xv.z, xv.w};
        const int tb = t0 + j * 4;
        float o[4], mf[4];
#pragma unroll
        for (int k = 0; k < 4; ++k) {
            int fl = loc[j * 4 + k];
            fl = (tex > fl) ? tex : fl;                    // full-row last start <= t
            int thr = tb + k - m;                          // window lower bound
            thr = (thr > 0) ? thr : 0;                     // clamp (matches reference)
            const bool mk = (fl >= thr);
            o[k]  = mk ? mval : xa[k];
            mf[k] = mk ? 1.0f : 0.0f;
        }
        const float4 ov = {o[0], o[1], o[2], o[3]};
        const float4 mv = {mf[0], mf[1], mf[2], mf[3]};
        orow[j] = ov;
        mrow[j] = mv;
    }
}

extern "C" void kernel_launch(void* const* d_in, const int* in_sizes, int n_in,
                              void* d_out, int out_size, void* d_ws, size_t ws_size,
                              hipStream_t stream) {
    const float* x    = (const float*)d_in[0];
    const int*   idx  = (const int*)d_in[1];
    const int*   msz  = (const int*)d_in[2];
    const int*   mval = (const int*)d_in[3];

    const int n    = in_sizes[0];        // B*S*T
    const int S    = in_sizes[2];        // 128
    const int rows = n / T_LEN;          // B*S

    float* out  = (float*)d_out;
    float* mout = out + n;               // tuple output #2 (mask) as float

    RandomSegmentMask_kernel<<<rows, BLK, 0, stream>>>(x, idx, msz, mval, out, mout, S);
}